// ScaledDotProductAttention_86517821216328
// MI455X (gfx1250) — compile-verified
//
#include <hip/hip_runtime.h>

typedef float v2f __attribute__((ext_vector_type(2)));
typedef float v8f __attribute__((ext_vector_type(8)));

#define SEQ 1024
#define DIM 64
#define BH  64           // B*H = 4*16
#define NEG_INF_F (-1000000000.0f)
#define INV_TEMP 0.125f  // 1/8

// ---------------------------------------------------------------------------
// CDNA5 async memory->LDS copy (no VGPR round trip, tracked by ASYNCcnt).
// LDS byte address = low 32 bits of the generic shared-memory pointer
// ---------------------------------------------------------------------------
__device__ __forceinline__ void async_copy_b128(const float* gsrc, float* ldst) {
    unsigned lds_addr = (unsigned)(unsigned long long)(uintptr_t)ldst;
    unsigned long long gaddr = (unsigned long long)(uintptr_t)gsrc;
    asm volatile("global_load_async_to_lds_b128 %0, %1, off"
                 :: "v"(lds_addr), "v"(gaddr) : "memory");
}

template <int N>
__device__ __forceinline__ void wait_async() {
#if __has_builtin(__builtin_amdgcn_s_wait_asynccnt)
    __builtin_amdgcn_s_wait_asynccnt(N);
#else
    asm volatile("s_wait_asynccnt %0" :: "i"(N) : "memory");
#endif
}

// ---------------------------------------------------------------------------
// Kernel 1: raw masked scores (written to attn region of d_out) + online
// per-COLUMN softmax stats (softmax is over the query axis!).
// Grid: (8 col-strips, 64 bh). Block: 256 threads = 8 waves.
// Wave w owns columns [strip*128 + w*16, +16) -> stats stay wave-private.
// Q tiles are double-buffered in LDS via async DMA so the copy of tile qt+1
// overlaps the WMMA work on tile qt.
// ---------------------------------------------------------------------------
__global__ __launch_bounds__(256)
void attn_scores_stats_kernel(const float* __restrict__ q,
                              const float* __restrict__ k,
                              const int*   __restrict__ mask,
                              float* __restrict__ attn_raw,
                              float* __restrict__ stats /* [BH][SEQ][2] */)
{
    __shared__ float qs[2][16 * DIM];  // double-buffered 16xD Q tiles (8 KB)

    const int bh   = blockIdx.y;
    const int wave = threadIdx.x >> 5;
    const int lane = threadIdx.x & 31;
    const int half = lane >> 4;     // 0: K/M 0..1 side, 1: K/M 2..3 side
    const int ln   = lane & 15;
    const int col0 = blockIdx.x * 128 + wave * 16;

    const float* Q = q    + (size_t)bh * SEQ * DIM;
    const float* K = k    + (size_t)bh * SEQ * DIM;
    const int*   M = mask + (size_t)bh * SEQ * SEQ;
    float*       A = attn_raw + (size_t)bh * SEQ * SEQ;

    // B operand (K^T tile) is invariant across q tiles: preload 16 fragments.
    // B[4x16] step s: lane holds K[col0+ln][4s + half*2 + {0,1}]
    v2f bfrag[16];
#pragma unroll
    for (int s = 0; s < 16; ++s)
        bfrag[s] = *(const v2f*)(K + (size_t)(col0 + ln) * DIM + 4 * s + half * 2);

    float m = -1e30f;   // running column max
    float l = 0.0f;     // running column sum of exp

    const int row  = threadIdx.x >> 4;       // Q tile staging mapping
    const int col4 = (threadIdx.x & 15) * 4;

    // prologue: issue async copy of tile 0 into buffer 0
    async_copy_b128(Q + (size_t)row * DIM + col4, &qs[0][row * DIM + col4]);

    for (int qt = 0; qt < SEQ / 16; ++qt) {
        const int buf = qt & 1;
        if (qt + 1 < SEQ / 16) {
            // issue next tile into the other buffer (safe: last iteration's
            // trailing barrier guarantees nobody still reads it)
            async_copy_b128(Q + (size_t)((qt + 1) * 16 + row) * DIM + col4,
                            &qs[buf ^ 1][row * DIM + col4]);
            wait_async<1>();   // async ops retire in order: tile qt has landed
        } else {
            wait_async<0>();
        }
        __syncthreads();

        v8f c = {};
#pragma unroll
        for (int s = 0; s < 16; ++s) {
            // A[16x4] step s: lane holds Q[q0+ln][4s + half*2 + {0,1}]
            v2f a = *(const v2f*)(&qs[buf][ln * DIM + 4 * s + half * 2]);
            c = __builtin_amdgcn_wmma_f32_16x16x4_f32(
                    false, a, false, bfrag[s], (short)0, c, false, false);
        }

        // scale + mask, write raw scores, update online column stats
        const int q0 = qt * 16;
        float sv[8];
        float tmax = -1e30f;
#pragma unroll
        for (int r = 0; r < 8; ++r) {
            const int qq = q0 + r + half * 8;      // C/D: row r (lanes<16) / r+8
            const int mm = M[(size_t)qq * SEQ + col0 + ln];
            const float sc = (mm == 0) ? NEG_INF_F : c[r] * INV_TEMP;
            sv[r] = sc;
            A[(size_t)qq * SEQ + col0 + ln] = sc;
            tmax = fmaxf(tmax, sc);
        }
        // combine the two lane-halves of each column (rows 0-7 with 8-15)
        tmax = fmaxf(tmax, __shfl_xor(tmax, 16, 32));
        const float nm = fmaxf(m, tmax);
        float psum = 0.0f;
#pragma unroll
        for (int r = 0; r < 8; ++r) psum += __expf(sv[r] - nm);
        psum += __shfl_xor(psum, 16, 32);
        l = l * __expf(m - nm) + psum;
        m = nm;

        __syncthreads();   // all waves done with qs[buf] before it is refilled
    }

    if (half == 0) {
        float* st = stats + ((size_t)bh * SEQ + col0 + ln) * 2;
        st[0] = m;
        st[1] = l;
    }
}

// ---------------------------------------------------------------------------
// Kernel 2: normalize attn in-place (exp(s-m)/l with per-COLUMN stats) and
// compute out = attn @ V fused (full k=1024 reduction inside the block).
// V is async-staged into LDS in double-buffered 64-row chunks (2 x 16 KB).
// Grid: (8 q-strips, 64 bh). Block: 256 threads = 8 waves.
// ---------------------------------------------------------------------------
#define VCHUNK 64
#define VCOPIES ((VCHUNK * DIM) / (256 * 4))   // async b128 per thread = 4

__global__ __launch_bounds__(256)
void attn_norm_out_kernel(const float* __restrict__ v,
                          const float* __restrict__ stats,
                          float* __restrict__ attn,   // raw in, normalized out
                          float* __restrict__ out)
{
    __shared__ float2 sml[SEQ];                // (m, l) per column, 8 KB
    __shared__ float  vs[2][VCHUNK * DIM];     // double-buffered V chunks, 32 KB

    const int bh   = blockIdx.y;
    const int wave = threadIdx.x >> 5;
    const int lane = threadIdx.x & 31;
    const int half = lane >> 4;
    const int ln   = lane & 15;
    const int q0   = blockIdx.x * 128 + wave * 16;

    for (int i = threadIdx.x; i < SEQ; i += 256)
        sml[i] = *(const float2*)(stats + ((size_t)bh * SEQ + i) * 2);

    const float* V = v    + (size_t)bh * SEQ * DIM;
    float*       A = attn + (size_t)bh * SEQ * SEQ;

    v8f c0 = {}, c1 = {}, c2 = {}, c3 = {};

    // prologue: issue async copy of chunk 0 into buffer 0
#pragma unroll
    for (int i = 0; i < VCOPIES; ++i) {
        const int idx = (threadIdx.x + i * 256) * 4;
        async_copy_b128(V + idx, &vs[0][idx]);
    }

    for (int ch = 0; ch < SEQ / VCHUNK; ++ch) {
        const int buf = ch & 1;
        if (ch + 1 < SEQ / VCHUNK) {
            const float* vsrc = V + (size_t)(ch + 1) * VCHUNK * DIM;
#pragma unroll
            for (int i = 0; i < VCOPIES; ++i) {
                const int idx = (threadIdx.x + i * 256) * 4;
                async_copy_b128(vsrc + idx, &vs[buf ^ 1][idx]);
            }
            wait_async<VCOPIES>();  // in-order: chunk ch's 4 copies have landed
        } else {
            wait_async<0>();
        }
        __syncthreads();

        for (int sl = 0; sl < VCHUNK / 4; ++sl) {
            const int kl = 4 * sl + half * 2;        // k-row within chunk
            const int kx = ch * VCHUNK + kl;         // global attn column
            float* ap = A + (size_t)(q0 + ln) * SEQ + kx;
            const v2f raw = *(const v2f*)ap;
            const float2 ml0 = sml[kx];
            const float2 ml1 = sml[kx + 1];
            v2f a;
            a.x = __expf(raw.x - ml0.x) / ml0.y;
            a.y = __expf(raw.y - ml1.x) / ml1.y;
            *(v2f*)ap = a;                           // write normalized attn

            // B operands from LDS: b.x = V[kl][d0+ln], b.y = V[kl+1][d0+ln]
            const float* vp = &vs[buf][kl * DIM + ln];
            v2f b0, b1, b2, b3;
            b0.x = vp[0];       b0.y = vp[DIM];
            b1.x = vp[16];      b1.y = vp[16 + DIM];
            b2.x = vp[32];      b2.y = vp[32 + DIM];
            b3.x = vp[48];      b3.y = vp[48 + DIM];

            c0 = __builtin_amdgcn_wmma_f32_16x16x4_f32(false, a, false, b0, (short)0, c0, false, false);
            c1 = __builtin_amdgcn_wmma_f32_16x16x4_f32(false, a, false, b1, (short)0, c1, false, false);
            c2 = __builtin_amdgcn_wmma_f32_16x16x4_f32(false, a, false, b2, (short)0, c2, false, false);
            c3 = __builtin_amdgcn_wmma_f32_16x16x4_f32(false, a, false, b3, (short)0, c3, false, false);
        }
        __syncthreads();   // all waves done with vs[buf] before it is refilled
    }

    float* O = out + ((size_t)bh * SEQ + q0) * DIM;
#pragma unroll
    for (int r = 0; r < 8; ++r) {
        const int qq = r + half * 8;
        float* orow = O + (size_t)qq * DIM + ln;
        orow[0]  = c0[r];
        orow[16] = c1[r];
        orow[32] = c2[r];
        orow[48] = c3[r];
    }
}

// ---------------------------------------------------------------------------
extern "C" void kernel_launch(void* const* d_in, const int* in_sizes, int n_in,
                              void* d_out, int out_size, void* d_ws, size_t ws_size,
                              hipStream_t stream) {
    const float* q    = (const float*)d_in[0];
    const float* k    = (const float*)d_in[1];
    const float* v    = (const float*)d_in[2];
    const int*   mask = (const int*)d_in[3];

    float* out  = (float*)d_out;                              // [BH][SEQ][DIM]
    float* attn = (float*)d_out + (size_t)BH * SEQ * DIM;     // [BH][SEQ][SEQ]
    float* stats = (float*)d_ws;                              // [BH][SEQ][2]

    dim3 block(256);
    dim3 grid1(8, BH);   // 8 column strips of 128
    attn_scores_stats_kernel<<<grid1, block, 0, stream>>>(q, k, mask, attn, stats);

    dim3 grid2(8, BH);   // 8 query strips of 128
    attn_norm_out_kernel<<<grid2, block, 0, stream>>>(v, stats, attn, out);
}